// CapsDecorelationNormalization2_89060441849996
// MI455X (gfx1250) — compile-verified
//
#include <hip/hip_runtime.h>
#include <hip/hip_bf16.h>

typedef __attribute__((ext_vector_type(2))) float v2f;
typedef __attribute__((ext_vector_type(8))) float v8f;
typedef __attribute__((ext_vector_type(4))) int   v4i;
typedef __attribute__((address_space(1))) v4i*    gv4i_ptr;  // global int4*
typedef __attribute__((address_space(3))) v4i*    lv4i_ptr;  // LDS int4*

#define CIN   32
#define BATCH 128
#define SPC   8192                   // COUT*ATOM samples per (b, channel)
#define NTOT  1048576                // total samples
#define TS    128                    // samples per LDS chunk (stats)
#define LS    132                    // padded LDS row stride (conflict-free)
#define NBLK_STATS 512
#define CHUNKS_PER_BLK 16            // 512 blocks * 16 * 128 = 1M samples
#define PART_STRIDE 1056             // 1024 (M) + 32 (s) floats per block
#define NBLK_APPLY 1024
#define TILES_PER_WAVE 8             // 1024 blk * 8 waves * 8 tiles * 16 = 1M

#if __has_builtin(__builtin_amdgcn_global_load_async_to_lds_b128)
#define ASYNC_LDS 1
#else
#define ASYNC_LDS 0
#endif

__device__ __forceinline__ void wait_async0() {
#if __has_builtin(__builtin_amdgcn_s_wait_asynccnt)
    __builtin_amdgcn_s_wait_asynccnt(0);
#else
    asm volatile("s_wait_asynccnt 0x0" ::: "memory");
#endif
}

// Stage one thread's 64B slice of a chunk into LDS (async if available).
__device__ __forceinline__ void stage_chunk(const float* g, float* l) {
#if ASYNC_LDS
    gv4i_ptr gp = (gv4i_ptr)g;
    lv4i_ptr lp = (lv4i_ptr)l;
    __builtin_amdgcn_global_load_async_to_lds_b128(gp, lp, 0, 0);
    __builtin_amdgcn_global_load_async_to_lds_b128(gp, lp, 16, 0);
    __builtin_amdgcn_global_load_async_to_lds_b128(gp, lp, 32, 0);
    __builtin_amdgcn_global_load_async_to_lds_b128(gp, lp, 48, 0);
#else
    const float4* s4 = (const float4*)g;
    float4* d4 = (float4*)l;
    d4[0] = s4[0]; d4[1] = s4[1]; d4[2] = s4[2]; d4[3] = s4[3];
#endif
}

// ---------------------------------------------------------------------------
// Kernel 1: raw statistics M = X^T X and channel sums, double-buffered LDS,
// V_WMMA_F32_16X16X4_F32, deterministic fixed-order reductions.
// ---------------------------------------------------------------------------
__global__ __launch_bounds__(256) void stats_kernel(const float* __restrict__ x,
                                                    float* __restrict__ part) {
    __shared__ float tile[2][CIN * LS];   // 2 x 32 x 132 floats (33.8 KB)
    __shared__ float wacc[8 * 1024];      // per-wave 32x32 spills (32 KB)
    __shared__ float csh[8][64];

    const int tid  = threadIdx.x;
    const int lane = tid & 31;
    const int wv   = tid >> 5;
    const int hf   = lane >> 4;
    const int sn   = lane & 15;

    const int b     = blockIdx.x >> 2;
    const int tbase = (blockIdx.x & 3) * (CHUNKS_PER_BLK * TS);
    const int ch    = tid >> 3;           // channel this thread stages
    const int soff  = (tid & 7) * 16;     // 16 consecutive floats
    const float* gbase = x + (size_t)b * CIN * SPC + (size_t)ch * SPC + tbase + soff;

    v8f acc[2][2];
    for (int ti = 0; ti < 2; ++ti)
        for (int tj = 0; tj < 2; ++tj)
            for (int r = 0; r < 8; ++r) acc[ti][tj][r] = 0.f;
    float cs0 = 0.f, cs1 = 0.f;

    stage_chunk(gbase, &tile[0][ch * LS + soff]);   // prologue: chunk 0

    for (int chunk = 0; chunk < CHUNKS_PER_BLK; ++chunk) {
#if ASYNC_LDS
        wait_async0();                    // this wave's fills for cur buf done
#endif
        __syncthreads();                  // all fills visible; prev reads done
        if (chunk + 1 < CHUNKS_PER_BLK)   // overlap next fill with compute
            stage_chunk(gbase + (chunk + 1) * TS,
                        &tile[(chunk + 1) & 1][ch * LS + soff]);

        const float* tl = tile[chunk & 1];
        const int ks0 = wv * 16;          // wave's 16-sample K slice
        for (int ks = 0; ks < 4; ++ks) {
            const int scol = ks0 + ks * 4 + 2 * hf;
            v2f f0, f1;                   // channel-tile fragments (A == B)
            f0.x = tl[sn * LS + scol];
            f0.y = tl[sn * LS + scol + 1];
            f1.x = tl[(16 + sn) * LS + scol];
            f1.y = tl[(16 + sn) * LS + scol + 1];
            cs0 += f0.x + f0.y;           // channel sums for free
            cs1 += f1.x + f1.y;
            acc[0][0] = __builtin_amdgcn_wmma_f32_16x16x4_f32(false, f0, false, f0,
                         (short)0, acc[0][0], false, false);
            acc[0][1] = __builtin_amdgcn_wmma_f32_16x16x4_f32(false, f0, false, f1,
                         (short)0, acc[0][1], false, false);
            acc[1][0] = __builtin_amdgcn_wmma_f32_16x16x4_f32(false, f1, false, f0,
                         (short)0, acc[1][0], false, false);
            acc[1][1] = __builtin_amdgcn_wmma_f32_16x16x4_f32(false, f1, false, f1,
                         (short)0, acc[1][1], false, false);
        }
    }

    // spill wave tiles (D layout: VGPR r -> M = hf*8+r, lane -> N = sn)
    for (int ti = 0; ti < 2; ++ti)
        for (int tj = 0; tj < 2; ++tj)
            for (int r = 0; r < 8; ++r) {
                const int i = ti * 16 + hf * 8 + r;
                const int j = tj * 16 + sn;
                wacc[wv * 1024 + i * 32 + j] = acc[ti][tj][r];
            }
    csh[wv][lane]      = cs0;             // partial for channel (lane&15)
    csh[wv][32 + lane] = cs1;             // partial for channel 16+(lane&15)
    __syncthreads();

    float* pout = part + (size_t)blockIdx.x * PART_STRIDE;
    for (int e = tid; e < 1024; e += 256) {
        float s = 0.f;
        for (int w = 0; w < 8; ++w) s += wacc[w * 1024 + e];
        pout[e] = s;
    }
    if (tid < 32) {
        float s = 0.f;
        if (tid < 16)
            for (int w = 0; w < 8; ++w) s += csh[w][tid] + csh[w][tid + 16];
        else
            for (int w = 0; w < 8; ++w) s += csh[w][16 + tid] + csh[w][32 + tid];
        pout[1024 + tid] = s;
    }
}

// ---------------------------------------------------------------------------
// Kernel 2: deterministic cross-block reduction of partials.
// ---------------------------------------------------------------------------
__global__ void reduce_kernel(const float* __restrict__ part,
                              float* __restrict__ Mout,
                              float* __restrict__ sout) {
    const int idx = blockIdx.x * blockDim.x + threadIdx.x;
    if (idx >= PART_STRIDE) return;
    float s = 0.f;
    for (int bb = 0; bb < NBLK_STATS; ++bb) s += part[(size_t)bb * PART_STRIDE + idx];
    if (idx < 1024) Mout[idx] = s;
    else            sout[idx - 1024] = s;
}

// ---------------------------------------------------------------------------
// Kernel 3: Newton-Schulz inverse-sqrt (32x32), one wave; folds mean/gamma/
// beta into Wg and off so the apply pass is a pure fused GEMM.
// ---------------------------------------------------------------------------
__global__ __launch_bounds__(32) void newton_kernel(const float* __restrict__ Msum,
                                                    const float* __restrict__ ssum,
                                                    const float* __restrict__ gamma,
                                                    const float* __restrict__ beta,
                                                    float* __restrict__ Wg,
                                                    float* __restrict__ offv) {
    __shared__ float sg[32][33], p[32][33], t1[32][33], t2[32][33];
    __shared__ float meanv[32];
    __shared__ float trsh;
    const int i = threadIdx.x;
    const float Nf = (float)NTOT;

    meanv[i] = ssum[i] / Nf;
    __syncthreads();
    const float invn1 = 1.0f / (Nf - 1.0f);
    for (int j = 0; j < 32; ++j)
        sg[i][j] = (Msum[i * 32 + j] - Nf * meanv[i] * meanv[j]) * invn1;
    __syncthreads();
    if (i == 0) {
        float tr = 0.f;
        for (int k = 0; k < 32; ++k) tr += sg[k][k];
        trsh = tr;
    }
    __syncthreads();
    const float invtr = 1.0f / trsh;
    for (int j = 0; j < 32; ++j) sg[i][j] *= invtr;
    for (int j = 0; j < 32; ++j) p[i][j] = (i == j) ? 1.0f : 0.0f;
    __syncthreads();

    for (int it = 0; it < 5; ++it) {
        for (int j = 0; j < 32; ++j) { float a = 0.f;
            for (int k = 0; k < 32; ++k) a += p[i][k] * p[k][j];   t1[i][j] = a; }
        __syncthreads();
        for (int j = 0; j < 32; ++j) { float a = 0.f;
            for (int k = 0; k < 32; ++k) a += t1[i][k] * p[k][j];  t2[i][j] = a; }
        __syncthreads();
        for (int j = 0; j < 32; ++j) { float a = 0.f;
            for (int k = 0; k < 32; ++k) a += t2[i][k] * sg[k][j]; t1[i][j] = a; }
        __syncthreads();
        for (int j = 0; j < 32; ++j) p[i][j] = 0.5f * (3.0f * p[i][j] - t1[i][j]);
        __syncthreads();
    }

    const float rs = rsqrtf(trsh);
    for (int j = 0; j < 32; ++j) Wg[i * 32 + j] = p[i][j] * rs * gamma[j];
    float a = 0.f;
    for (int k = 0; k < 32; ++k) a += meanv[k] * p[k][i];
    offv[i] = beta[i] - a * rs * gamma[i];
}

// ---------------------------------------------------------------------------
// Kernel 4: out = X * Wg + off.  Channels on WMMA M axis, samples on N axis
// -> contiguous 64B runs per half-wave for both x loads and out stores.
// Non-temporal stores keep x resident in L2 (192 MB) for this second pass.
// ---------------------------------------------------------------------------
__global__ __launch_bounds__(256) void apply_kernel(const float* __restrict__ x,
                                                    const float* __restrict__ Wg,
                                                    const float* __restrict__ offv,
                                                    float* __restrict__ out) {
    const int lane = threadIdx.x & 31;
    const int wv   = threadIdx.x >> 5;
    const int hf   = lane >> 4;
    const int sn   = lane & 15;
    const int gw   = blockIdx.x * 8 + wv;

    // A fragments: A[M=ci_local, K=k] = Wg[k][ct*16 + ci_local]
    v2f af[2][8];
    for (int ct = 0; ct < 2; ++ct)
        for (int kk = 0; kk < 8; ++kk) {
            const int k0 = kk * 4 + 2 * hf;
            const int ci = ct * 16 + sn;
            af[ct][kk].x = Wg[k0 * 32 + ci];
            af[ct][kk].y = Wg[(k0 + 1) * 32 + ci];
        }
    v8f cinit[2];
    for (int ct = 0; ct < 2; ++ct)
        for (int r = 0; r < 8; ++r) cinit[ct][r] = offv[ct * 16 + hf * 8 + r];

    for (int it = 0; it < TILES_PER_WAVE; ++it) {
        const int tIdx  = gw * TILES_PER_WAVE + it;   // 65536 sample tiles
        const int nbase = tIdx * 16;
        const int b     = nbase >> 13;
        const int t0    = nbase & (SPC - 1);
        const float* xb = x + (size_t)b * CIN * SPC + t0 + sn;

        v8f a0 = cinit[0], a1 = cinit[1];
        for (int kk = 0; kk < 8; ++kk) {
            const int k0 = kk * 4 + 2 * hf;
            v2f bf;                        // B[K=k, N=sample]
            bf.x = xb[(size_t)k0 * SPC];
            bf.y = xb[(size_t)(k0 + 1) * SPC];
            a0 = __builtin_amdgcn_wmma_f32_16x16x4_f32(false, af[0][kk], false, bf,
                     (short)0, a0, false, false);
            a1 = __builtin_amdgcn_wmma_f32_16x16x4_f32(false, af[1][kk], false, bf,
                     (short)0, a1, false, false);
        }

        float* ob = out + (size_t)b * CIN * SPC + t0 + sn;
        for (int r = 0; r < 8; ++r) {
            __builtin_nontemporal_store(a0[r], ob + (size_t)(hf * 8 + r) * SPC);
            __builtin_nontemporal_store(a1[r], ob + (size_t)(16 + hf * 8 + r) * SPC);
        }
    }
}

// ---------------------------------------------------------------------------
extern "C" void kernel_launch(void* const* d_in, const int* in_sizes, int n_in,
                              void* d_out, int out_size, void* d_ws, size_t ws_size,
                              hipStream_t stream) {
    const float* x     = (const float*)d_in[0];
    const float* gamma = (const float*)d_in[1];
    const float* beta  = (const float*)d_in[2];
    float* out = (float*)d_out;

    float* ws   = (float*)d_ws;
    float* part = ws;                                     // 512 * 1056 floats
    float* Mfin = ws + (size_t)NBLK_STATS * PART_STRIDE;  // 1024
    float* sfin = Mfin + 1024;                            // 32
    float* Wg   = sfin + 32;                              // 1024
    float* offv = Wg + 1024;                              // 32

    stats_kernel<<<NBLK_STATS, 256, 0, stream>>>(x, part);
    reduce_kernel<<<(PART_STRIDE + 255) / 256, 256, 0, stream>>>(part, Mfin, sfin);
    newton_kernel<<<1, 32, 0, stream>>>(Mfin, sfin, gamma, beta, Wg, offv);
    apply_kernel<<<NBLK_APPLY, 256, 0, stream>>>(x, Wg, offv, out);
}